// DeepFusionNetV2_20538533609643
// MI455X (gfx1250) — compile-verified
//
#include <hip/hip_runtime.h>
#include <math.h>

typedef float v2f __attribute__((ext_vector_type(2)));
typedef float v8f __attribute__((ext_vector_type(8)));

// ---------------------------------------------------------------------------
// Haar LLL: [4,1,128,128,128] -> [4,1,64,64,64], 2x2x2 block sum * 0.5^1.5
// ---------------------------------------------------------------------------
__global__ __launch_bounds__(256) void wavelet_kernel(const float* __restrict__ in,
                                                      float* __restrict__ out) {
  int idx = blockIdx.x * 256 + threadIdx.x;            // 4*64^3 = 1,048,576
  int x = idx & 63, y = (idx >> 6) & 63, z = (idx >> 12) & 63, n = idx >> 18;
  const float* p = in + ((long)n << 21) + (z << 1) * 16384 + (y << 1) * 128 + (x << 1);
  float s = p[0] + p[1] + p[128] + p[129] + p[16384] + p[16385] + p[16512] + p[16513];
  out[idx] = s * 0.35355339059327373f;
}

// ---------------------------------------------------------------------------
// First VGG conv: Cin=1 -> Cout=32, 3x3x3 pad1, S=64, + BN + ReLU (direct)
// ---------------------------------------------------------------------------
__global__ __launch_bounds__(256) void conv0_kernel(const float* __restrict__ in,
                                                    const float* __restrict__ W,
                                                    const float* __restrict__ B,
                                                    const float* __restrict__ sc,
                                                    const float* __restrict__ sh,
                                                    float* __restrict__ out) {
  const int S = 64, S2 = 4096, S3 = 262144;
  int idx = blockIdx.x * 256 + threadIdx.x;            // 4*32*64^3
  int x = idx & 63, y = (idx >> 6) & 63, z = (idx >> 12) & 63;
  int co = (idx >> 18) & 31, n = idx >> 23;
  const float* w = W + co * 27;
  const float* src = in + n * S3;
  float sum = 0.f;
  #pragma unroll
  for (int dz = 0; dz < 3; dz++) {
    int zz = z + dz - 1;
    if ((unsigned)zz >= 64u) continue;
    #pragma unroll
    for (int dy = 0; dy < 3; dy++) {
      int yy = y + dy - 1;
      if ((unsigned)yy >= 64u) continue;
      #pragma unroll
      for (int dx = 0; dx < 3; dx++) {
        int xx = x + dx - 1;
        if ((unsigned)xx >= 64u) continue;
        sum += w[dz * 9 + dy * 3 + dx] * src[zz * S2 + yy * S + xx];
      }
    }
  }
  float v = (sum + B[co]) * sc[co] + sh[co];
  out[idx] = fmaxf(v, 0.f);
}

// ---------------------------------------------------------------------------
// 2x2x2 maxpool: [4,C,S,S,S] -> [4,C,S/2,S/2,S/2]
// ---------------------------------------------------------------------------
__global__ __launch_bounds__(256) void maxpool_kernel(const float* __restrict__ in,
                                                      float* __restrict__ out,
                                                      int C, int S) {
  int So = S >> 1, So3 = So * So * So;
  int total = 4 * C * So3;
  int idx = blockIdx.x * 256 + threadIdx.x;
  if (idx >= total) return;
  int x = idx % So, r = idx / So;
  int y = r % So; r /= So;
  int z = r % So; r /= So;
  int c = r % C, n = r / C;
  const float* p = in + (((n * C + c) * S + 2 * z) * S + 2 * y) * S + 2 * x;
  int S2 = S * S;
  float m = p[0];
  m = fmaxf(m, p[1]);
  m = fmaxf(m, p[S]); m = fmaxf(m, p[S + 1]);
  m = fmaxf(m, p[S2]); m = fmaxf(m, p[S2 + 1]);
  m = fmaxf(m, p[S2 + S]); m = fmaxf(m, p[S2 + S + 1]);
  out[idx] = m;
}

// ---------------------------------------------------------------------------
// Implicit-GEMM conv3d via V_WMMA_F32_16X16X4_F32, tap-major K ordering:
//   k' = tap*Cin + ci  (WMMA sums over K, so ordering is free).
//   Per tap: dz,dy,dx are wave-uniform scalars; bounds + base addresses
//   hoisted per tile; inner ci-loop is pure pointer stepping (+4*S3).
//   A element for row k' is W[co][ci*KVOL + tap] (stride-KVOL gather).
//   Wave tile: 16 co x 64 pos (4 accumulators sharing the A fragment).
//   Epilogue: (acc + bias)*bn_s + bn_t, optional ReLU.
// ---------------------------------------------------------------------------
template <int KS, bool RELU>
__global__ __launch_bounds__(256) void conv_wmma(const float* __restrict__ in,
                                                 const float* __restrict__ W,
                                                 const float* __restrict__ bias,
                                                 const float* __restrict__ bns,
                                                 const float* __restrict__ bnt,
                                                 float* __restrict__ out,
                                                 int Cin, int Cout, int S) {
  const int KVOL = KS * KS * KS;                       // 27 or 1
  const int K = Cin * KVOL;
  const int S2 = S * S, S3 = S2 * S;
  const int lane = threadIdx.x & 31;
  const int wave = threadIdx.x >> 5;
  const int chunk = blockIdx.x * 8 + wave;             // 64-position chunk
  const int co0 = blockIdx.y * 16;
  const int p0 = chunk * 64;
  const int col = lane & 15;                           // N column within tile
  const int kh = (lane >> 4) << 1;                     // K sub-row: 0 or 2
  const int P = (KS == 3) ? 1 : 0;

  int nt[4], zt[4], yt[4], xt[4];
  #pragma unroll
  for (int t = 0; t < 4; t++) {
    int p = p0 + t * 16;
    int n = p / S3, rem = p - n * S3;
    nt[t] = n;
    zt[t] = rem / S2; rem -= zt[t] * S2;
    yt[t] = rem / S;
    xt[t] = rem - yt[t] * S;
  }

  v8f acc[4] = {};
  const float* Abase = W + (co0 + col) * K + kh * KVOL;  // A: M = co0+(lane&15)

  for (int tap = 0; tap < KVOL; ++tap) {
    const int dz = (KS == 3) ? tap / 9 : 0;
    const int dy = (KS == 3) ? (tap / 3) % 3 : 0;
    const int dx = (KS == 3) ? tap % 3 : 0;

    // Per-tile hoisted bounds + clamped base pointers (loads stay
    // unconditional -> EXEC remains all-ones for WMMA; zeros via cndmask).
    bool okf[4];
    const float* pB[4];
    #pragma unroll
    for (int t = 0; t < 4; t++) {
      int zz = zt[t] + dz - P;
      int yy = yt[t] + dy - P;
      int xx = xt[t] + dx - P + col;
      bool ok = (unsigned)zz < (unsigned)S && (unsigned)yy < (unsigned)S &&
                (unsigned)xx < (unsigned)S;
      okf[t] = ok;
      int offp = ok ? (zz * S2 + yy * S + xx) : 0;
      pB[t] = in + (nt[t] * Cin + kh) * S3 + offp;
    }

    const float* Wp = Abase + tap;
    for (int ci = 0; ci < Cin; ci += 4) {
      v2f a;
      a.x = Wp[0];
      a.y = Wp[KVOL];
      Wp += 4 * KVOL;
      #pragma unroll
      for (int t = 0; t < 4; t++) {
        float b0 = pB[t][0];
        float b1 = pB[t][S3];
        pB[t] += 4 * S3;
        v2f b;
        b.x = okf[t] ? b0 : 0.f;
        b.y = okf[t] ? b1 : 0.f;
        acc[t] = __builtin_amdgcn_wmma_f32_16x16x4_f32(
            false, a, false, b, (short)0, acc[t], false, false);
      }
    }
  }

  // C layout: VGPR r, lane L -> M = r + (L<16?0:8), N = L&15
  const int mhi = (lane >> 4) << 3;
  #pragma unroll
  for (int t = 0; t < 4; t++) {
    int sp = zt[t] * S2 + yt[t] * S + xt[t] + col;
    #pragma unroll
    for (int r = 0; r < 8; r++) {
      int co = co0 + r + mhi;
      float v = (acc[t][r] + bias[co]) * bns[co] + bnt[co];
      if (RELU) v = fmaxf(v, 0.f);
      out[(nt[t] * Cout + co) * S3 + sp] = v;
    }
  }
}

// ---------------------------------------------------------------------------
// Gate: sigmoid(conv1x1([bf;hf])) and blend, fused.  S=16, C=128, K=256.
// K loop split into bf half (k<128) and hf half (k>=128), pointer-stepped.
// ---------------------------------------------------------------------------
__global__ __launch_bounds__(256) void gate_blend_kernel(const float* __restrict__ bf,
                                                         const float* __restrict__ hf,
                                                         const float* __restrict__ gw,
                                                         const float* __restrict__ gb,
                                                         float* __restrict__ out) {
  const int S3 = 4096;
  const int lane = threadIdx.x & 31;
  const int wave = threadIdx.x >> 5;
  const int chunk = blockIdx.x * 8 + wave;
  const int co0 = blockIdx.y * 16;
  const int p0 = chunk * 64;
  const int col = lane & 15;
  const int kh = (lane >> 4) << 1;

  int nt[4], spt[4];
  #pragma unroll
  for (int t = 0; t < 4; t++) {
    int p = p0 + t * 16;
    nt[t] = p >> 12;
    spt[t] = (p & 4095) + col;
  }

  v8f acc[4] = {};
  const float* Arow = gw + (co0 + col) * 256 + kh;

  const float* srcs[2] = {bf, hf};
  #pragma unroll
  for (int h = 0; h < 2; h++) {
    const float* Wp = Arow + h * 128;
    const float* pB[4];
    #pragma unroll
    for (int t = 0; t < 4; t++)
      pB[t] = srcs[h] + (nt[t] * 128 + kh) * S3 + spt[t];
    for (int ci = 0; ci < 128; ci += 4) {
      v2f a;
      a.x = Wp[0];
      a.y = Wp[1];
      Wp += 4;
      #pragma unroll
      for (int t = 0; t < 4; t++) {
        v2f b;
        b.x = pB[t][0];
        b.y = pB[t][S3];
        pB[t] += 4 * S3;
        acc[t] = __builtin_amdgcn_wmma_f32_16x16x4_f32(
            false, a, false, b, (short)0, acc[t], false, false);
      }
    }
  }

  const int mhi = (lane >> 4) << 3;
  #pragma unroll
  for (int t = 0; t < 4; t++) {
    #pragma unroll
    for (int r = 0; r < 8; r++) {
      int co = co0 + r + mhi;
      float g = 1.f / (1.f + expf(-(acc[t][r] + gb[co])));
      int off = (nt[t] * 128 + co) * S3 + spt[t];
      out[off] = g * bf[off] + (1.f - g) * hf[off];
    }
  }
}

// ---------------------------------------------------------------------------
// SE: channel means, tiny MLP, channel scale
// ---------------------------------------------------------------------------
__global__ __launch_bounds__(256) void se_mean_kernel(const float* __restrict__ x,
                                                      float* __restrict__ mean) {
  __shared__ float sm[256];
  const float* p = x + (long)blockIdx.x * 4096;
  float s = 0.f;
  for (int i = threadIdx.x; i < 4096; i += 256) s += p[i];
  sm[threadIdx.x] = s;
  __syncthreads();
  for (int st = 128; st > 0; st >>= 1) {
    if (threadIdx.x < st) sm[threadIdx.x] += sm[threadIdx.x + st];
    __syncthreads();
  }
  if (threadIdx.x == 0) mean[blockIdx.x] = sm[0] * (1.f / 4096.f);
}

__global__ __launch_bounds__(128) void se_fc_kernel(const float* __restrict__ mean,
                                                    const float* __restrict__ w1,
                                                    const float* __restrict__ w2,
                                                    float* __restrict__ scale) {
  __shared__ float yv[128];
  __shared__ float hid[8];
  int n = blockIdx.x, c = threadIdx.x;
  yv[c] = mean[n * 128 + c];
  __syncthreads();
  if (c < 8) {
    float s = 0.f;
    for (int i = 0; i < 128; i++) s += w1[c * 128 + i] * yv[i];
    hid[c] = fmaxf(s, 0.f);
  }
  __syncthreads();
  float s = 0.f;
  #pragma unroll
  for (int j = 0; j < 8; j++) s += w2[c * 8 + j] * hid[j];
  scale[n * 128 + c] = 1.f / (1.f + expf(-s));
}

__global__ __launch_bounds__(256) void se_apply_kernel(float* __restrict__ x,
                                                       const float* __restrict__ scale) {
  int idx = blockIdx.x * 256 + threadIdx.x;            // 4*128*4096
  int c = (idx >> 12) & 127, n = idx >> 19;
  x[idx] *= scale[n * 128 + c];
}

// ---------------------------------------------------------------------------
// Host launcher
// ---------------------------------------------------------------------------
extern "C" void kernel_launch(void* const* d_in, const int* in_sizes, int n_in,
                              void* d_out, int out_size, void* d_ws, size_t ws_size,
                              hipStream_t stream) {
  (void)in_sizes; (void)n_in; (void)out_size; (void)ws_size;
  auto I = [&](int i) { return (const float*)d_in[i]; };

  // d_in pytree order: b, h, brain[6x(w,b,s,t)]=2..25, eca[3x(w,b,s,t)]=26..37,
  // gate_b=38, gate_w=39, hipp[6x(w,b,s,t)]=40..63, se_w1=64, se_w2=65
  float* ws = (float*)d_ws;
  size_t off = 0;
  float* wl   = ws + off; off += 1048576;    // [4,1,64^3]
  float* buf0 = ws + off; off += 33554432;   // up to [4,32,64^3]
  float* buf1 = ws + off; off += 33554432;
  float* p1   = ws + off; off += 4194304;    // [4,32,32^3]
  float* p3   = ws + off; off += 1048576;    // [4,64,16^3]
  float* bfb  = ws + off; off += 2097152;    // [4,128,16^3]
  float* hfb  = ws + off; off += 2097152;
  float* fus  = ws + off; off += 2097152;
  float* e1   = ws + off; off += 2097152;
  float* e2   = ws + off; off += 2097152;
  float* mean = ws + off; off += 512;
  float* scal = ws + off; off += 512;

  auto run_branch = [&](const float* x, int b, float* F) {
    wavelet_kernel<<<4096, 256, 0, stream>>>(x, wl);
    conv0_kernel<<<131072, 256, 0, stream>>>(wl, I(b), I(b + 1), I(b + 2), I(b + 3), buf0);
    conv_wmma<3, true><<<dim3(2048, 2), 256, 0, stream>>>(
        buf0, I(b + 4), I(b + 5), I(b + 6), I(b + 7), buf1, 32, 32, 64);
    maxpool_kernel<<<16384, 256, 0, stream>>>(buf1, p1, 32, 64);
    conv_wmma<3, true><<<dim3(256, 4), 256, 0, stream>>>(
        p1, I(b + 8), I(b + 9), I(b + 10), I(b + 11), buf0, 32, 64, 32);
    conv_wmma<3, true><<<dim3(256, 4), 256, 0, stream>>>(
        buf0, I(b + 12), I(b + 13), I(b + 14), I(b + 15), buf1, 64, 64, 32);
    maxpool_kernel<<<4096, 256, 0, stream>>>(buf1, p3, 64, 32);
    conv_wmma<3, true><<<dim3(32, 8), 256, 0, stream>>>(
        p3, I(b + 16), I(b + 17), I(b + 18), I(b + 19), buf0, 64, 128, 16);
    conv_wmma<3, true><<<dim3(32, 8), 256, 0, stream>>>(
        buf0, I(b + 20), I(b + 21), I(b + 22), I(b + 23), F, 128, 128, 16);
  };

  run_branch(I(0), 2, bfb);    // brain
  run_branch(I(1), 40, hfb);   // hipp

  gate_blend_kernel<<<dim3(32, 8), 256, 0, stream>>>(bfb, hfb, I(39), I(38), fus);

  se_mean_kernel<<<512, 256, 0, stream>>>(fus, mean);
  se_fc_kernel<<<4, 128, 0, stream>>>(mean, I(64), I(65), scal);
  se_apply_kernel<<<8192, 256, 0, stream>>>(fus, scal);

  // ECA: 1x1 -> 3x3 -> 1x1, BN only (no ReLU)
  conv_wmma<1, false><<<dim3(32, 8), 256, 0, stream>>>(
      fus, I(26), I(27), I(28), I(29), e1, 128, 128, 16);
  conv_wmma<3, false><<<dim3(32, 8), 256, 0, stream>>>(
      e1, I(30), I(31), I(32), I(33), e2, 128, 128, 16);
  conv_wmma<1, false><<<dim3(32, 8), 256, 0, stream>>>(
      e2, I(34), I(35), I(36), I(37), (float*)d_out, 128, 128, 16);
}